// VolumeRenderer_42314017800362
// MI455X (gfx1250) — compile-verified
//
#include <hip/hip_runtime.h>
#include <hip/hip_bf16.h>
#include <stdint.h>

// Volume rendering: B rays x N samples, C feature channels.
// One wave32 per ray; lane l owns samples 4l..4l+3 (all loads coalesced b128).
// CDNA5 path: per-wave TDM tensor_load_to_lds streams the ray's feature slice
// (384 f32) into LDS asynchronously (TENSORcnt) while the wave computes the
// transmittance scan from depths/densities.

#define NS    128   // samples per ray
#define CF    3     // feature channels
#define WAVES 8     // waves (rays) per block
#define TPB   256
#define FAR_DELTA 1e10f

#if defined(__has_builtin)
#  if __has_builtin(__builtin_amdgcn_tensor_load_to_lds) && \
      __has_builtin(__builtin_amdgcn_s_wait_tensorcnt)    && \
      __has_builtin(__builtin_amdgcn_readfirstlane)
#    define USE_TDM 1
#  endif
#endif
#ifndef USE_TDM
#  define USE_TDM 0
#endif

typedef unsigned int u32x4 __attribute__((ext_vector_type(4)));
typedef int          i32x4 __attribute__((ext_vector_type(4)));
typedef int          i32x8 __attribute__((ext_vector_type(8)));

__global__ __launch_bounds__(TPB) void volrender_kernel(
    const float* __restrict__ lengths,   // [B, NS]
    const float* __restrict__ density,   // [B, NS]
    const float* __restrict__ feature,   // [B, NS, CF]
    float* __restrict__ feat_out,        // [B, CF]
    float* __restrict__ depth_out,       // [B]
    int B)
{
#if USE_TDM
  __shared__ float lds_feat[WAVES * NS * CF];  // 12 KB
#endif
  const int wave = threadIdx.x >> 5;
  const int lane = threadIdx.x & 31;
  const int ray  = blockIdx.x * WAVES + wave;
  if (ray >= B) return;

#if USE_TDM
  {
    // ---- Build Tensor DMA Descriptor (D#) for a 1D 384-element f32 tile ----
    const int sray = __builtin_amdgcn_readfirstlane(ray);
    // LDS aperture: low 32 bits of generic pointer == LDS byte offset.
    unsigned lbase  = (unsigned)(uintptr_t)(void*)&lds_feat[0];
    unsigned ldsoff = (unsigned)__builtin_amdgcn_readfirstlane(
        (int)(lbase + (unsigned)(wave * (NS * CF * 4))));
    uint64_t ga = (uint64_t)(uintptr_t)(feature + (size_t)sray * (NS * CF));

    u32x4 g0;
    g0[0] = 1u;                                        // count=1, user D#
    g0[1] = ldsoff;                                    // lds_addr (bytes)
    g0[2] = (unsigned)ga;                              // global_addr[31:0]
    g0[3] = ((unsigned)(ga >> 32) & 0x01FFFFFFu)       // global_addr[56:32]
            | (2u << 30);                              // type=2 ("image")

    const unsigned E = (unsigned)(NS * CF);            // 384 elements
    i32x8 g1;
    g1[0] = (int)(2u << 16);                           // data_size=2 (4B); mask=0
    g1[1] = (int)((E & 0xFFFFu) << 16);                // tensor_dim0[15:0]
    g1[2] = (int)(1u << 16);                           // tensor_dim0[31:16]=0; tensor_dim1=1
    g1[3] = (int)(E << 16);                            // tensor_dim1 hi=0; tile_dim0=384
    g1[4] = 1;                                         // tile_dim1=1; tile_dim2=0
    g1[5] = (int)E;                                    // tensor_dim0_stride[31:0]
    g1[6] = (int)(E << 16);                            // stride0 hi=0; tensor_dim1_stride lo=384
    g1[7] = 0;                                         // tensor_dim1_stride hi
    i32x4 g2; g2[0] = 1; g2[1] = 1; g2[2] = 0; g2[3] = 0;   // tensor_dim2/3=1
    i32x4 g3; g3[0] = 0; g3[1] = 0; g3[2] = 0; g3[3] = 0;
    i32x8 g4; g4[0]=0; g4[1]=0; g4[2]=0; g4[3]=0; g4[4]=0; g4[5]=0; g4[6]=0; g4[7]=0; // VADDR4 group (unused)

    // 6-arg form (clang-23 / therock-10.0 lane): (g0,g1,g2,g3,g4,cpol)
    __builtin_amdgcn_tensor_load_to_lds(g0, g1, g2, g3, g4, 0); // TENSORcnt++
  }
#endif

  // ---- Coalesced b128 loads of this lane's 4 depths + 4 densities ----
  const float4 d4 = *(const float4*)(lengths + (size_t)ray * NS + lane * 4);
  const float4 s4 = *(const float4*)(density + (size_t)ray * NS + lane * 4);

  // deltas: next lane's first depth closes this lane's last interval
  const float nx = __shfl_down(d4.x, 1, 32);
  const float dl0 = d4.y - d4.x;
  const float dl1 = d4.z - d4.y;
  const float dl2 = d4.w - d4.z;
  const float dl3 = (lane == 31) ? FAR_DELTA : (nx - d4.w);

  // per-sample transmittance factors
  const float e0 = __expf(-s4.x * dl0);
  const float e1 = __expf(-s4.y * dl1);
  const float e2 = __expf(-s4.z * dl2);
  const float e3 = __expf(-s4.w * dl3);

  // exclusive prefix product across the wave (5-step wave32 scan)
  const float p01 = e0 * e1;
  float scan = p01 * e2 * e3;                  // local product of 4 factors
  #pragma unroll
  for (int off = 1; off < 32; off <<= 1) {
    float v = __shfl_up(scan, off, 32);
    if (lane >= off) scan *= v;
  }
  float excl = __shfl_up(scan, 1, 32);
  if (lane == 0) excl = 1.0f;

  const float T0 = excl;
  const float T1 = excl * e0;
  const float T2 = excl * p01;
  const float T3 = T2 * e2;

  const float w0 = T0 * (1.0f - e0);
  const float w1 = T1 * (1.0f - e1);
  const float w2 = T2 * (1.0f - e2);
  const float w3 = T3 * (1.0f - e3);

  // ---- Features: from LDS (TDM) or straight from global ----
#if USE_TDM
  __builtin_amdgcn_s_wait_tensorcnt(0);
  const float* fb = &lds_feat[wave * (NS * CF) + lane * 12];
  asm volatile("" : "+v"(fb) : : "memory");    // pin LDS reads after the wait
#else
  const float* fb = feature + (size_t)ray * (NS * CF) + lane * 12;
#endif
  const float4 fa = *(const float4*)(fb + 0);  // s0c0 s0c1 s0c2 s1c0
  const float4 fm = *(const float4*)(fb + 4);  // s1c1 s1c2 s2c0 s2c1
  const float4 fc = *(const float4*)(fb + 8);  // s2c2 s3c0 s3c1 s3c2

  float a0 = w0 * fa.x + w1 * fa.w + w2 * fm.z + w3 * fc.y;
  float a1 = w0 * fa.y + w1 * fm.x + w2 * fm.w + w3 * fc.z;
  float a2 = w0 * fa.z + w1 * fm.y + w2 * fc.x + w3 * fc.w;
  float ad = w0 * d4.x + w1 * d4.y + w2 * d4.z + w3 * d4.w;

  // ---- wave32 tree reduction ----
  #pragma unroll
  for (int off = 16; off >= 1; off >>= 1) {
    a0 += __shfl_down(a0, off, 32);
    a1 += __shfl_down(a1, off, 32);
    a2 += __shfl_down(a2, off, 32);
    ad += __shfl_down(ad, off, 32);
  }
  if (lane == 0) {
    feat_out[(size_t)ray * CF + 0] = a0;
    feat_out[(size_t)ray * CF + 1] = a1;
    feat_out[(size_t)ray * CF + 2] = a2;
    depth_out[ray] = ad;
  }
}

extern "C" void kernel_launch(void* const* d_in, const int* in_sizes, int n_in,
                              void* d_out, int out_size, void* d_ws, size_t ws_size,
                              hipStream_t stream) {
  (void)n_in; (void)out_size; (void)d_ws; (void)ws_size;
  const float* lengths = (const float*)d_in[0];   // [B,N,1]
  const float* density = (const float*)d_in[1];   // [B,N,1]
  const float* feature = (const float*)d_in[2];   // [B,N,C]
  const int B = in_sizes[0] / NS;

  float* out       = (float*)d_out;
  float* feat_out  = out;                  // [B,C]
  float* depth_out = out + (size_t)B * CF; // [B,1]

  const int blocks = (B + WAVES - 1) / WAVES;
  volrender_kernel<<<blocks, TPB, 0, stream>>>(lengths, density, feature,
                                               feat_out, depth_out, B);
}